// GCN_82592221102294
// MI455X (gfx1250) — compile-verified
//
#include <hip/hip_runtime.h>
#include <math.h>

#define NN 100000
#define NE 1200000
#define DIM 64
#define NG 64
#define EPSBN 1e-5f

typedef float v2f __attribute__((ext_vector_type(2)));
typedef float v8f __attribute__((ext_vector_type(8)));

// ---------------- utility kernels ----------------

__global__ void zero_f32(float* __restrict__ p, int n) {
  int i = blockIdx.x * blockDim.x + threadIdx.x;
  if (i < n) p[i] = 0.0f;
}

__global__ void deg_kernel(const int* __restrict__ dst, float* __restrict__ deg, int E) {
  int e = blockIdx.x * blockDim.x + threadIdx.x;
  if (e < E) atomicAdd(&deg[dst[e]], 1.0f);
}

__global__ void dis_kernel(float* __restrict__ deg, int n) {
  int i = blockIdx.x * blockDim.x + threadIdx.x;
  if (i < n) deg[i] = rsqrtf(deg[i] + 1.0f);  // deg includes +1 self loop
}

// ---------------- WMMA GEMM: H = X * W  (X: n x 64, W: 64 x 64) ----------------
// 128 threads = 4 waves per block; each wave computes a 16x64 output slab with
// V_WMMA_F32_16X16X4_F32 (full f32, matches reference dtype).
// W is staged in LDS k-pair-interleaved: sW2[p][n] = {W[2p][n], W[2p+1][n]}
// so each B fragment is one aligned ds_load_b64 into an even VGPR pair.

__global__ __launch_bounds__(128) void gemm_wmma_f32(const float* __restrict__ X,
                                                     const float* __restrict__ W,
                                                     float* __restrict__ H,
                                                     int nrows) {
  __shared__ v2f sW2[32 * DIM];  // 16 KB, [k-pair p][col n]

  const int tid = threadIdx.x;
  // stage W k-pair interleaved (rows are coalesced per 64-thread group)
  for (int i = tid; i < 32 * DIM; i += 128) {
    int p = i >> 6;       // k-pair index 0..31
    int n = i & 63;       // column
    v2f w;
    w.x = W[(2 * p + 0) * DIM + n];
    w.y = W[(2 * p + 1) * DIM + n];
    sW2[i] = w;
  }
  __syncthreads();

  const int wave = tid >> 5;
  const int lane = tid & 31;
  const int m = lane & 15;            // row within tile (A) / col within tile (B,C)
  const int hi = lane >> 4;           // 0: lanes 0-15, 1: lanes 16-31
  const int khalf = hi * 2;           // K=0,1 vs K=2,3 within each k-step
  const int r0 = (blockIdx.x * 4 + wave) * 16;
  if (r0 >= nrows) return;  // wave-uniform (nrows % 16 == 0) -> EXEC stays all-1s

  // Preload all A fragments for this wave's 16 rows: 16 k-steps x float2
  v2f a[16];
  const float* xrow = X + (size_t)(r0 + m) * DIM;
#pragma unroll
  for (int kk = 0; kk < 16; ++kk) {
    a[kk] = *(const v2f*)(xrow + kk * 4 + khalf);  // 8B aligned (khalf even)
  }

  // 4 independent accumulator tiles (16 cols each) -> 4 independent WMMA chains
  v8f acc[4];
#pragma unroll
  for (int t = 0; t < 4; ++t) acc[t] = (v8f){0.f, 0.f, 0.f, 0.f, 0.f, 0.f, 0.f, 0.f};

#pragma unroll
  for (int kk = 0; kk < 16; ++kk) {
    const v2f* brow = sW2 + (kk * 2 + hi) * DIM + m;  // k-pair row for this lane half
#pragma unroll
    for (int t = 0; t < 4; ++t) {
      v2f b = brow[t * 16];  // single ds_load_b64, even-aligned pair
      acc[t] = __builtin_amdgcn_wmma_f32_16x16x4_f32(
          /*neg_a=*/false, a[kk], /*neg_b=*/false, b,
          /*c_mod=*/(short)0, acc[t], /*reuse_a=*/false, /*reuse_b=*/false);
    }
  }

  // Store: C/D layout -> VGPR v holds row (v + 8*hi), col = lane&15
  const int rbase = r0 + (hi << 3);
#pragma unroll
  for (int t = 0; t < 4; ++t) {
#pragma unroll
    for (int v = 0; v < 8; ++v) {
      H[(size_t)(rbase + v) * DIM + t * 16 + m] = acc[t][v];
    }
  }
}

// ---------------- edge scatter: OUT[dst] += dis[src]*dis[dst] * H[src] ----------------
// 16 threads per edge, float4 gather; atomics stay resident in the 192MB L2.

__global__ void edge_kernel(const int* __restrict__ src, const int* __restrict__ dst,
                            const float* __restrict__ dis, const float* __restrict__ H,
                            float* __restrict__ OUT, int E) {
  int idx = blockIdx.x * blockDim.x + threadIdx.x;
  if (idx >= E * (DIM / 4)) return;
  int e = idx >> 4;
  int d = (idx & 15) * 4;
  int s = src[e];
  int t = dst[e];
  float w = dis[s] * dis[t];
  const float4 hv = *(const float4*)(H + (size_t)s * DIM + d);
  float* o = OUT + (size_t)t * DIM + d;
  atomicAdd(o + 0, w * hv.x);
  atomicAdd(o + 1, w * hv.y);
  atomicAdd(o + 2, w * hv.z);
  atomicAdd(o + 3, w * hv.w);
}

// ---------------- fused self-loop + bias + BN(eval) + ReLU (float4) ----------------

__global__ void post_bn_relu(float* __restrict__ AGG, const float* __restrict__ H,
                             const float* __restrict__ dis, const float* __restrict__ bias,
                             const float* __restrict__ g, const float* __restrict__ be,
                             const float* __restrict__ rm, const float* __restrict__ rv,
                             int n) {
  int idx = blockIdx.x * blockDim.x + threadIdx.x;
  if (idx >= n * (DIM / 4)) return;
  int node = idx >> 4;
  int d = (idx & 15) * 4;
  float di = dis[node];
  float d2 = di * di;
  float4 aggv = *(float4*)(AGG + (size_t)idx * 4);
  float4 hv = *(const float4*)(H + (size_t)idx * 4);
  float4 bv = *(const float4*)(bias + d);
  float4 gv = *(const float4*)(g + d);
  float4 bev = *(const float4*)(be + d);
  float4 rmv = *(const float4*)(rm + d);
  float4 rvv = *(const float4*)(rv + d);
  float4 o;
  o.x = fmaxf((aggv.x + d2 * hv.x + bv.x - rmv.x) * rsqrtf(rvv.x + EPSBN) * gv.x + bev.x, 0.f);
  o.y = fmaxf((aggv.y + d2 * hv.y + bv.y - rmv.y) * rsqrtf(rvv.y + EPSBN) * gv.y + bev.y, 0.f);
  o.z = fmaxf((aggv.z + d2 * hv.z + bv.z - rmv.z) * rsqrtf(rvv.z + EPSBN) * gv.z + bev.z, 0.f);
  o.w = fmaxf((aggv.w + d2 * hv.w + bv.w - rmv.w) * rsqrtf(rvv.w + EPSBN) * gv.w + bev.w, 0.f);
  *(float4*)(AGG + (size_t)idx * 4) = o;
}

__global__ void post_plain(float* __restrict__ AGG, const float* __restrict__ H,
                           const float* __restrict__ dis, const float* __restrict__ bias,
                           int n) {
  int idx = blockIdx.x * blockDim.x + threadIdx.x;
  if (idx >= n * (DIM / 4)) return;
  int node = idx >> 4;
  int d = (idx & 15) * 4;
  float di = dis[node];
  float d2 = di * di;
  float4 aggv = *(float4*)(AGG + (size_t)idx * 4);
  float4 hv = *(const float4*)(H + (size_t)idx * 4);
  float4 bv = *(const float4*)(bias + d);
  float4 o;
  o.x = aggv.x + d2 * hv.x + bv.x;
  o.y = aggv.y + d2 * hv.y + bv.y;
  o.z = aggv.z + d2 * hv.z + bv.z;
  o.w = aggv.w + d2 * hv.w + bv.w;
  *(float4*)(AGG + (size_t)idx * 4) = o;
}

// ---------------- global mean pool ----------------

__global__ void pool_accum(const float* __restrict__ H, const int* __restrict__ batch,
                           float* __restrict__ hg, float* __restrict__ counts, int n) {
  int idx = blockIdx.x * blockDim.x + threadIdx.x;
  if (idx >= n * (DIM / 4)) return;
  int node = idx >> 4;
  int d = (idx & 15) * 4;
  int gph = batch[node];
  float4 hv = *(const float4*)(H + (size_t)node * DIM + d);
  float* o = hg + (size_t)gph * DIM + d;
  atomicAdd(o + 0, hv.x);
  atomicAdd(o + 1, hv.y);
  atomicAdd(o + 2, hv.z);
  atomicAdd(o + 3, hv.w);
  if ((idx & 15) == 0) atomicAdd(&counts[gph], 1.0f);
}

__global__ void pool_div(float* __restrict__ hg, const float* __restrict__ counts) {
  int idx = blockIdx.x * blockDim.x + threadIdx.x;
  if (idx < NG * DIM) hg[idx] /= fmaxf(counts[idx >> 6], 1.0f);
}

// ---------------- launch ----------------

extern "C" void kernel_launch(void* const* d_in, const int* in_sizes, int n_in,
                              void* d_out, int out_size, void* d_ws, size_t ws_size,
                              hipStream_t stream) {
  (void)in_sizes; (void)n_in; (void)out_size; (void)ws_size;

  const float* x   = (const float*)d_in[0];
  const int* eidx  = (const int*)d_in[1];
  const int* src   = eidx;          // edge_index[0]
  const int* dstp  = eidx + NE;     // edge_index[1]
  const int* batch = (const int*)d_in[2];
  const float* W1 = (const float*)d_in[3];
  const float* W2 = (const float*)d_in[4];
  const float* W3 = (const float*)d_in[5];
  const float* b1 = (const float*)d_in[6];
  const float* b2 = (const float*)d_in[7];
  const float* b3 = (const float*)d_in[8];
  const float* g1 = (const float*)d_in[9];
  const float* be1 = (const float*)d_in[10];
  const float* rm1 = (const float*)d_in[11];
  const float* rv1 = (const float*)d_in[12];
  const float* g2 = (const float*)d_in[13];
  const float* be2 = (const float*)d_in[14];
  const float* rm2 = (const float*)d_in[15];
  const float* rv2 = (const float*)d_in[16];

  float* ws = (float*)d_ws;
  float* bufH   = ws;                         // N*D
  float* bufAgg = ws + (size_t)NN * DIM;      // N*D
  float* dis    = ws + (size_t)2 * NN * DIM;  // N
  float* counts = dis + NN;                   // G

  float* h_out = (float*)d_out;               // N*D (first tuple element)
  float* hg    = h_out + (size_t)NN * DIM;    // G*D (second tuple element)

  const int ND = NN * DIM;
  const int gridND = (ND + 255) / 256;
  const int gridND4 = (NN * (DIM / 4) + 255) / 256;
  const int gridED4 = (NE * (DIM / 4) + 255) / 256;
  const int gemmBlocks = ((NN / 16) + 3) / 4;  // 4 row-tiles (waves) per block

  // degrees -> dis = rsqrt(deg+1)
  zero_f32<<<(NN + 255) / 256, 256, 0, stream>>>(dis, NN);
  deg_kernel<<<(NE + 255) / 256, 256, 0, stream>>>(dstp, dis, NE);
  dis_kernel<<<(NN + 255) / 256, 256, 0, stream>>>(dis, NN);

  // ---- layer 1 ----
  gemm_wmma_f32<<<gemmBlocks, 128, 0, stream>>>(x, W1, bufH, NN);
  zero_f32<<<gridND, 256, 0, stream>>>(bufAgg, ND);
  edge_kernel<<<gridED4, 256, 0, stream>>>(src, dstp, dis, bufH, bufAgg, NE);
  post_bn_relu<<<gridND4, 256, 0, stream>>>(bufAgg, bufH, dis, b1, g1, be1, rm1, rv1, NN);

  // ---- layer 2 ----
  gemm_wmma_f32<<<gemmBlocks, 128, 0, stream>>>(bufAgg, W2, bufH, NN);
  zero_f32<<<gridND, 256, 0, stream>>>(bufAgg, ND);
  edge_kernel<<<gridED4, 256, 0, stream>>>(src, dstp, dis, bufH, bufAgg, NE);
  post_bn_relu<<<gridND4, 256, 0, stream>>>(bufAgg, bufH, dis, b2, g2, be2, rm2, rv2, NN);

  // ---- layer 3 (into d_out) ----
  gemm_wmma_f32<<<gemmBlocks, 128, 0, stream>>>(bufAgg, W3, bufH, NN);
  zero_f32<<<gridND, 256, 0, stream>>>(h_out, ND);
  edge_kernel<<<gridED4, 256, 0, stream>>>(src, dstp, dis, bufH, h_out, NE);
  post_plain<<<gridND4, 256, 0, stream>>>(h_out, bufH, dis, b3, NN);

  // ---- global mean pool ----
  zero_f32<<<(NG * DIM + 255) / 256, 256, 0, stream>>>(hg, NG * DIM);
  zero_f32<<<1, 256, 0, stream>>>(counts, NG);
  pool_accum<<<gridND4, 256, 0, stream>>>(h_out, batch, hg, counts, NN);
  pool_div<<<(NG * DIM + 255) / 256, 256, 0, stream>>>(hg, counts);
}